// AI4CFD_4733053960390
// MI455X (gfx1250) — compile-verified
//
#include <hip/hip_runtime.h>

// ---------------------------------------------------------------------------
// Reference: 100 Richardson iterations of u <- u - (A(u) - u_old),
//   A(u) = u - DT*L(u),  L(u) = RE*conv(u,w1)-UB*conv(u,w2)-UB*conv(u,w3)
// Folded: residual = u + conv3x3(u, w_eff) + c0 - u_old
//   w_eff = DT*(UB*(w2+w3) - RE*w1),  c0 = DT*(UB*(b2+b3) - RE*b1)
// Norm tracking / convergence flag / conditional update per the scan body.
//
// Conv tile = 16x16 per wave, computed with 15x V_WMMA_F32_16X16X4_F32:
//   Y = sum_{dy=0..2} U_dy (16 x 20) * Wband_dy (20 x 16)
// Accumulator seeded with (u + c0 - u_old) so WMMA chain yields the residual.
// ---------------------------------------------------------------------------

typedef __attribute__((ext_vector_type(2))) float v2f;
typedef __attribute__((ext_vector_type(8))) float v8f;

#define NPIX (1024 * 1024)
#define W_IMG 1024
#define NTILE_X 64              // 1024/16
#define NBLOCKS 512             // 4096 tiles / 8 waves per block
#define MAX_ITER 100

struct Ctl {
    unsigned counter;
    int converged;
    float scale;
    float final_res;
};

__device__ __forceinline__ float bandv(int d, float w0, float w1, float w2) {
    // Wband[k][j] = w_eff_row[k-j] for 0<=k-j<=2 else 0  (cndmask chain, no spill)
    float r = 0.0f;
    r = (d == 0) ? w0 : r;
    r = (d == 1) ? w1 : r;
    r = (d == 2) ? w2 : r;
    return r;
}

__global__ void __launch_bounds__(256)
ai4cfd_init(float* __restrict__ u, const float* __restrict__ u_old,
            float* __restrict__ weff, Ctl* ctl,
            const float* __restrict__ w1, const float* __restrict__ w2,
            const float* __restrict__ w3, const float* __restrict__ b1,
            const float* __restrict__ b2, const float* __restrict__ b3, int n) {
    const float DT = 0.05f, RE = 0.0f, UBc = 1.0f;
    int i = blockIdx.x * blockDim.x + threadIdx.x;
    int stride = gridDim.x * blockDim.x;
    for (int k = i; k < n; k += stride) u[k] = u_old[k];
    if (i == 0) {
        #pragma unroll
        for (int t = 0; t < 9; ++t)
            weff[t] = DT * (UBc * (w2[t] + w3[t]) - RE * w1[t]);
        weff[9] = DT * (UBc * (b2[0] + b3[0]) - RE * b1[0]);
        ctl->counter = 0u;
        ctl->converged = 0;
        ctl->scale = 1.0f;
        ctl->final_res = 0.0f;
    }
}

// One iteration's residual + deterministic norm reduction + last-block finalize.
__global__ void __launch_bounds__(256)
ai4cfd_conv(const float* __restrict__ u, const float* __restrict__ u_old,
            float* __restrict__ R, const float* __restrict__ weff,
            float* __restrict__ partials, Ctl* ctl) {
    __shared__ float tile[8][18][20];   // per-wave 18x20 halo tile (zero padded)
    __shared__ float red[256];
    __shared__ int isLast;

    const int tid = threadIdx.x;
    const int w = tid >> 5;             // wave in block (wave32)
    const int lane = tid & 31;
    const int wid = blockIdx.x * 8 + w; // global tile id
    const int r0 = (wid >> 6) * 16;     // tile row origin
    const int c0 = (wid & (NTILE_X - 1)) * 16;

    // ---- stage u halo tile into LDS (zero outside image) ----
    for (int idx = lane; idx < 360; idx += 32) {
        int tr = idx / 20;
        int tc = idx - tr * 20;
        int gr = r0 - 1 + tr;
        int gc = c0 - 1 + tc;
        float v = 0.0f;
        if ((unsigned)gr < 1024u && (unsigned)gc < 1024u)
            v = u[gr * W_IMG + gc];
        tile[w][tr][tc] = v;
    }
    __syncthreads();

    const int m = lane & 15;            // A row / B col / D col
    const int hi = lane >> 4;           // half-wave selector
    const int kb = hi * 2;              // K sub-offset for A/B fragments

    // stencil coefficients (uniform)
    float wf[9];
    #pragma unroll
    for (int t = 0; t < 9; ++t) wf[t] = weff[t];
    const float c0s = weff[9];

    // ---- seed accumulator: c = u + c0 - u_old  (C/D layout: lane=N, vgpr=Mmod8)
    v8f acc;
    #pragma unroll
    for (int vv = 0; vv < 8; ++vv) {
        int M = vv + hi * 8;
        float uc = tile[w][M + 1][m + 1];
        float uo = u_old[(r0 + M) * W_IMG + (c0 + m)];
        acc[vv] = uc + c0s - uo;
    }

    // ---- 15x WMMA f32 16x16x4: banded-matrix conv ----
    #pragma unroll
    for (int dy = 0; dy < 3; ++dy) {
        float wr0 = wf[dy * 3 + 0], wr1 = wf[dy * 3 + 1], wr2 = wf[dy * 3 + 2];
        #pragma unroll
        for (int kc = 0; kc < 5; ++kc) {
            int kk = kc * 4 + kb;
            v2f a = { tile[w][m + dy][kk], tile[w][m + dy][kk + 1] };
            v2f b = { bandv(kk - m, wr0, wr1, wr2),
                      bandv(kk + 1 - m, wr0, wr1, wr2) };
            acc = __builtin_amdgcn_wmma_f32_16x16x4_f32(
                false, a, false, b, (short)0, acc, false, false);
        }
    }

    // ---- write residual, accumulate ||r||^2 ----
    float s = 0.0f;
    #pragma unroll
    for (int vv = 0; vv < 8; ++vv) {
        int M = vv + hi * 8;
        float r = acc[vv];
        R[(r0 + M) * W_IMG + (c0 + m)] = r;
        s += r * r;
    }

    // deterministic block tree reduction
    red[tid] = s;
    __syncthreads();
    #pragma unroll
    for (int off = 128; off > 0; off >>= 1) {
        if (tid < off) red[tid] += red[tid + off];
        __syncthreads();
    }
    if (tid == 0) {
        partials[blockIdx.x] = red[0];
        __threadfence();
        unsigned t = atomicAdd(&ctl->counter, 1u);
        isLast = (t == (unsigned)(NBLOCKS - 1)) ? 1 : 0;
    }
    __syncthreads();

    if (isLast) {
        // last block: deterministic sum of 512 partials, finalize flags
        float p = 0.0f;
        for (int k = tid; k < NBLOCKS; k += 256)
            p += ((volatile float*)partials)[k];
        red[tid] = p;
        __syncthreads();
        #pragma unroll
        for (int off = 128; off > 0; off >>= 1) {
            if (tid < off) red[tid] += red[tid + off];
            __syncthreads();
        }
        if (tid == 0) {
            float nrm = sqrtf(red[0]);
            int was = ctl->converged;
            if (!was) ctl->final_res = nrm;
            int now = was | ((nrm < 1e-6f) ? 1 : 0);
            ctl->converged = now;
            ctl->scale = now ? 0.0f : 1.0f;
            ctl->counter = 0u;      // rearm for next iteration
            __threadfence();
        }
    }
}

__global__ void __launch_bounds__(256)
ai4cfd_update(float* __restrict__ u, const float* __restrict__ R,
              const Ctl* __restrict__ ctl, int n) {
    float s = ctl->scale;
    int i = blockIdx.x * blockDim.x + threadIdx.x;
    int stride = gridDim.x * blockDim.x;
    for (int k = i; k < n; k += stride) u[k] -= s * R[k];
}

__global__ void __launch_bounds__(256)
ai4cfd_final(const float* __restrict__ u, const Ctl* __restrict__ ctl,
             float* __restrict__ out, int n) {
    int i = blockIdx.x * blockDim.x + threadIdx.x;
    int stride = gridDim.x * blockDim.x;
    for (int k = i; k < n; k += stride) out[k] = u[k];
    if (i == 0) out[n] = ctl->final_res;
}

extern "C" void kernel_launch(void* const* d_in, const int* in_sizes, int n_in,
                              void* d_out, int out_size, void* d_ws, size_t ws_size,
                              hipStream_t stream) {
    const float* u_old = (const float*)d_in[0];
    const float* w1 = (const float*)d_in[1];
    const float* w2 = (const float*)d_in[2];
    const float* w3 = (const float*)d_in[3];
    const float* b1 = (const float*)d_in[4];
    const float* b2 = (const float*)d_in[5];
    const float* b3 = (const float*)d_in[6];
    float* out = (float*)d_out;
    const int n = NPIX;

    // workspace layout (all 64B aligned): u | weff(16f) | ctl | partials(512f)
    float* u = (float*)d_ws;
    float* weff = u + n;
    Ctl* ctl = (Ctl*)((char*)d_ws + (size_t)n * 4 + 64);
    float* partials = (float*)((char*)d_ws + (size_t)n * 4 + 128);
    float* R = out;   // residual scratch lives in d_out[0..n); overwritten at end

    ai4cfd_init<<<1024, 256, 0, stream>>>(u, u_old, weff, ctl,
                                          w1, w2, w3, b1, b2, b3, n);
    for (int it = 0; it < MAX_ITER; ++it) {
        ai4cfd_conv<<<NBLOCKS, 256, 0, stream>>>(u, u_old, R, weff, partials, ctl);
        ai4cfd_update<<<2048, 256, 0, stream>>>(u, R, ctl, n);
    }
    ai4cfd_final<<<2048, 256, 0, stream>>>(u, ctl, out, n);
}